// Conv1d_NN_spatial_28664611734129
// MI455X (gfx1250) — compile-verified
//
#include <hip/hip_runtime.h>

// Problem constants (from reference)
#define B_   8
#define C_   64
#define N_   4096
#define M_   1024
#define KNN  3
#define OC_  64
#define CK_  192   // C_ * KNN

typedef __attribute__((ext_vector_type(2))) float v2f;
typedef __attribute__((ext_vector_type(8))) float v8f;
typedef __attribute__((ext_vector_type(4))) unsigned int u32x4;
typedef __attribute__((ext_vector_type(4))) int i32x4;
typedef __attribute__((ext_vector_type(8))) int i32x8;

// ---------------------------------------------------------------------------
// Kernel 0: norm_y[b,m] = sum_c y[b,c,m]^2   (8192 values, trivial)
// ---------------------------------------------------------------------------
__global__ void normy_kernel(const float* __restrict__ y, float* __restrict__ ny) {
  int t = blockIdx.x * blockDim.x + threadIdx.x;   // 0 .. B_*M_-1
  int b = t >> 10;
  int m = t & (M_ - 1);
  const float* yp = y + (size_t)b * C_ * M_ + m;
  float s = 0.f;
#pragma unroll
  for (int c = 0; c < C_; ++c) { float v = yp[(size_t)c * M_]; s = fmaf(v, v, s); }
  ny[t] = s;
}

// ---------------------------------------------------------------------------
// Branchless top-3 (smallest) insertion: 3 v_cmp + 10 v_cndmask, no branches.
// Strict '<' => earliest (lowest) index wins ties; per-lane m's arrive ascending.
// ---------------------------------------------------------------------------
__device__ __forceinline__ void ins3(float s, int m,
    float& d0, int& i0, float& d1, int& i1, float& d2, int& i2) {
  bool c0 = s < d0, c1 = s < d1, c2 = s < d2;
  float nd2 = c1 ? d1 : (c2 ? s : d2);
  int   ni2 = c1 ? i1 : (c2 ? m : i2);
  float nd1 = c0 ? d0 : (c1 ? s : d1);
  int   ni1 = c0 ? i0 : (c1 ? m : i1);
  float nd0 = c0 ? s : d0;
  int   ni0 = c0 ? m : i0;
  d0 = nd0; i0 = ni0; d1 = nd1; i1 = ni1; d2 = nd2; i2 = ni2;
}

// lexicographic (value, index) version for the cross-half merge
__device__ __forceinline__ void ins3lex(float s, int m,
    float& d0, int& i0, float& d1, int& i1, float& d2, int& i2) {
  bool c0 = (s < d0) || (s == d0 && m < i0);
  bool c1 = (s < d1) || (s == d1 && m < i1);
  bool c2 = (s < d2) || (s == d2 && m < i2);
  float nd2 = c1 ? d1 : (c2 ? s : d2);
  int   ni2 = c1 ? i1 : (c2 ? m : i2);
  float nd1 = c0 ? d0 : (c1 ? s : d1);
  int   ni1 = c0 ? i0 : (c1 ? m : i1);
  float nd0 = c0 ? s : d0;
  int   ni0 = c0 ? m : i0;
  d0 = nd0; i0 = ni0; d1 = nd1; i1 = ni1; d2 = nd2; i2 = ni2;
}

// ---------------------------------------------------------------------------
// TDM: DMA one 16(m) x 64(c) fp32 y-tile into LDS (compact [c][m] layout).
//   D# group0: count=1 | lds_addr | global_addr | type=2
//   D# group1: data_size=4B, tensor_dim0=M, tensor_dim1=C,
//              tile_dim0=16, tile_dim1=64, tensor_dim0_stride=M
// Issued by one wave; tracked by TENSORcnt.
// ---------------------------------------------------------------------------
__device__ __forceinline__ void tdm_load_tile(const float* gsrc, float* lds_dst) {
  unsigned long long ga = (unsigned long long)(uintptr_t)gsrc;
  unsigned lo32 = (unsigned)(uintptr_t)lds_dst;   // low 32 bits = LDS byte offset
  u32x4 g0;
  g0[0] = 1u;                                   // count = 1 valid descriptor
  g0[1] = lo32;                                 // lds_addr
  g0[2] = (unsigned)ga;                         // global_addr[31:0]
  g0[3] = (unsigned)(ga >> 32) | 0x80000000u;   // global_addr[56:32] | type=2<<30
  i32x8 g1;
  g1[0] = 0x00020000;        // wg_mask=0, data_size=2 (4 bytes), no flags
  g1[1] = 0x04000000;        // tensor_dim0 = 1024 (bits 79:48 -> low16 here)
  g1[2] = 0x00400000;        // tensor_dim0 hi16=0 | tensor_dim1 = 64 (low16)
  g1[3] = 0x00100000;        // tensor_dim1 hi16=0 | tile_dim0 = 16
  g1[4] = 64;                // tile_dim1 = 64, tile_dim2 = 0
  g1[5] = 1024;              // tensor_dim0_stride = 1024 (low 32)
  g1[6] = 0;                 // stride hi16 = 0 | tensor_dim1_stride lo = 0
  g1[7] = 0;
  i32x4 gz = {0, 0, 0, 0};   // 2-D tile: groups 2/3 unused
#if __has_include(<hip/amd_detail/amd_gfx1250_TDM.h>)
  i32x8 gz8 = {0, 0, 0, 0, 0, 0, 0, 0};
  __builtin_amdgcn_tensor_load_to_lds(g0, g1, gz, gz, gz8, 0);
#else
  __builtin_amdgcn_tensor_load_to_lds(g0, g1, gz, gz, 0);
#endif
}

// ---------------------------------------------------------------------------
// Fused kernel: per wave = one (b, 16-wide n-tile).
//   Stage B: dot GEMM via V_WMMA_F32_16X16X4_F32 on TDM-staged y tiles,
//            fused score + branchless top-3.
//   Stage C: gather 3 neighbor columns, grouped conv via WMMA, bias+relu.
// Block = 128 threads (4 waves); y tiles double-buffered in LDS via TDM.
// ---------------------------------------------------------------------------
__launch_bounds__(128)
__global__ void knn_conv_kernel(const float* __restrict__ x,
                                const float* __restrict__ y,
                                const float* __restrict__ w,
                                const float* __restrict__ bias,
                                const float* __restrict__ ny,
                                float* __restrict__ out) {
  __shared__ float sY[2][C_ * 16];   // [buf][c*16 + m] compact, 4 KB each

  const int tid  = threadIdx.x;
  const int lane = tid & 31;
  const int wv   = tid >> 5;
  const int ml   = lane & 15;   // n column (B/D), m row (A), o row (conv A)
  const int hl   = lane >> 4;   // half-wave select (K sub-pair)

  const int nb = (int)blockIdx.x & (N_ / 64 - 1);
  const int b  = (int)blockIdx.x / (N_ / 64);
  const int n0 = nb * 64 + wv * 16;

  const float* xb  = x  + (size_t)b * C_ * N_;
  const float* yb  = y  + (size_t)b * C_ * M_;
  const float* nyb = ny + (size_t)b * M_;

  // ---- preload loop-invariant B operand: bx[t] = x[c = 4t+2hl+{0,1}][n0+ml]
  v2f bx[16];
#pragma unroll
  for (int t = 0; t < 16; ++t) {
    int c = 4 * t + 2 * hl;
    bx[t].x = xb[(size_t)c       * N_ + n0 + ml];
    bx[t].y = xb[(size_t)(c + 1) * N_ + n0 + ml];
  }

  float d0 = __builtin_inff(), d1 = __builtin_inff(), d2 = __builtin_inff();
  int   i0 = 0, i1 = 0, i2 = 0;

  // ---- TDM-stage m-tile 0 into buffer 0 (wave 0 issues; TDM ignores EXEC)
  if (wv == 0) tdm_load_tile(yb, &sY[0][0]);
  __builtin_amdgcn_s_wait_tensorcnt((short)0);
  __syncthreads();

  for (int tile = 0; tile < 64; ++tile) {
    const int buf = tile & 1;
    // stage next tile into the other buffer while we compute this one
    if (wv == 0 && tile + 1 < 64) {
      tdm_load_tile(yb + (tile + 1) * 16, &sY[buf ^ 1][0]);
    }

    // D[m][n] = sum_c y[c][m] * x[c][n] ; 16 K-steps of 4 over C=64
    v8f acc = {0.f, 0.f, 0.f, 0.f, 0.f, 0.f, 0.f, 0.f};
    const float* yt = &sY[buf][0];
#pragma unroll
    for (int t = 0; t < 16; ++t) {
      int c0 = 4 * t + 2 * hl;
      v2f a;
      a.x = yt[c0 * 16 + ml];
      a.y = yt[(c0 + 1) * 16 + ml];
      acc = __builtin_amdgcn_wmma_f32_16x16x4_f32(false, a, false, bx[t],
                                                  (short)0, acc, false, false);
    }

    // score = norm_y[m] - 2*dot ; lane's rows m = tile*16 + 8*hl + v
    const int m0 = tile * 16 + 8 * hl;
    float4 nyA = *(const float4*)(nyb + m0);
    float4 nyB = *(const float4*)(nyb + m0 + 4);
    float nv[8] = {nyA.x, nyA.y, nyA.z, nyA.w, nyB.x, nyB.y, nyB.z, nyB.w};
#pragma unroll
    for (int v = 0; v < 8; ++v) {
      float s = fmaf(-2.f, acc[v], nv[v]);
      ins3(s, m0 + v, d0, i0, d1, i1, d2, i2);
    }

    __builtin_amdgcn_s_wait_tensorcnt((short)0);  // no-op for waves 1..3
    __syncthreads();
  }

  // ---- merge the two half-wave candidate sets (rows 0-7 vs 8-15 of each tile)
  {
    float e0 = __shfl_xor(d0, 16, 32); int j0 = __shfl_xor(i0, 16, 32);
    float e1 = __shfl_xor(d1, 16, 32); int j1 = __shfl_xor(i1, 16, 32);
    float e2 = __shfl_xor(d2, 16, 32); int j2 = __shfl_xor(i2, 16, 32);
    ins3lex(e0, j0, d0, i0, d1, i1, d2, i2);
    ins3lex(e1, j1, d0, i0, d1, i1, d2, i2);
    ins3lex(e2, j2, d0, i0, d1, i1, d2, i2);
  }
  // now every lane holds the 3 nearest indices for column n = n0 + ml

  // ---- Stage C: out[o, n] = relu( W[64x192] * patch[192 x 16] + bias )
  float* outb = out + (size_t)b * OC_ * N_;
#pragma unroll 1
  for (int ot = 0; ot < 4; ++ot) {
    const int o0 = ot * 16;
    v8f acc = {0.f, 0.f, 0.f, 0.f, 0.f, 0.f, 0.f, 0.f};
#pragma unroll
    for (int t = 0; t < 48; ++t) {
      const int k0 = 4 * t + 2 * hl;
      // A operand: conv weights, row o = o0 + ml, kdims (k0, k0+1) contiguous
      v2f aw = *(const v2f*)(w + (size_t)(o0 + ml) * CK_ + k0);
      // B operand: gathered neighbor values; kdim = c*3 + k
      v2f bp;
      {
        int c  = k0 / 3, kk = k0 % 3;
        int id = (kk == 0) ? i0 : ((kk == 1) ? i1 : i2);
        bp.x = xb[(size_t)c * N_ + id];
        int k1 = k0 + 1;
        c  = k1 / 3; kk = k1 % 3;
        id = (kk == 0) ? i0 : ((kk == 1) ? i1 : i2);
        bp.y = xb[(size_t)c * N_ + id];
      }
      acc = __builtin_amdgcn_wmma_f32_16x16x4_f32(false, aw, false, bp,
                                                  (short)0, acc, false, false);
    }
    // bias + relu + store: lane's rows o = o0 + 8*hl + v, col n = n0 + ml
    float4 bA = *(const float4*)(bias + o0 + 8 * hl);
    float4 bB = *(const float4*)(bias + o0 + 8 * hl + 4);
    float bb[8] = {bA.x, bA.y, bA.z, bA.w, bB.x, bB.y, bB.z, bB.w};
#pragma unroll
    for (int v = 0; v < 8; ++v) {
      float r = acc[v] + bb[v];
      r = (r > 0.f) ? r : 0.f;
      outb[(size_t)(o0 + 8 * hl + v) * N_ + n0 + ml] = r;
    }
  }
}

// ---------------------------------------------------------------------------
extern "C" void kernel_launch(void* const* d_in, const int* in_sizes, int n_in,
                              void* d_out, int out_size, void* d_ws, size_t ws_size,
                              hipStream_t stream) {
  (void)in_sizes; (void)n_in; (void)out_size; (void)ws_size;
  const float* x    = (const float*)d_in[0];
  const float* y    = (const float*)d_in[1];
  const float* w    = (const float*)d_in[2];
  const float* bias = (const float*)d_in[3];
  float* ny  = (float*)d_ws;          // B_*M_ floats = 32 KB scratch
  float* out = (float*)d_out;

  normy_kernel<<<(B_ * M_) / 256, 256, 0, stream>>>(y, ny);
  knn_conv_kernel<<<B_ * (N_ / 64), 128, 0, stream>>>(x, y, w, bias, ny, out);
}